// mHC_CMS_Block_80564996538489
// MI455X (gfx1250) — compile-verified
//
#include <hip/hip_runtime.h>
#include <math.h>

typedef __attribute__((ext_vector_type(2))) float v2f;
typedef __attribute__((ext_vector_type(4))) float f4;
typedef __attribute__((ext_vector_type(8))) float v8f;
typedef __attribute__((ext_vector_type(4))) int v4i;
typedef __attribute__((address_space(1))) v4i gv4i;   // global v4i32
typedef __attribute__((address_space(3))) v4i sv4i;   // LDS v4i32

#define DDIM 2048
#define SDIM 4
#define TOK 4                  // tokens per group (16 WMMA rows)
#define NTHREADS 256           // 8 waves (wave32)
#define NWAVES 8
#define KPW (DDIM / NWAVES)    // 256 K per wave
#define CHUNKS (KPW / 4)       // 64 WMMA 16x16x4 steps per wave
#define NGROUPS (2 * 4096 / TOK)   // 2048
#define GPW 4                  // groups per workgroup (double-buffer pipeline)
#define NWG (NGROUPS / GPW)    // 512 workgroups
#define RSIZE (TOK * SDIM * DDIM)  // 32768 floats per group

#if __has_builtin(__builtin_amdgcn_global_load_async_to_lds_b128)
#define HAVE_ASYNC_LDS 1
#else
#define HAVE_ASYNC_LDS 0
#endif

__device__ __forceinline__ void gload_lds128(const float* g, float* l) {
#if HAVE_ASYNC_LDS
  __builtin_amdgcn_global_load_async_to_lds_b128((gv4i*)g, (sv4i*)l, 0, 0);
#else
  *(f4*)l = *(const f4*)g;   // fallback: vmem load + ds store
#endif
}

__device__ __forceinline__ void wait_async_lds() {
#if HAVE_ASYNC_LDS
#if __has_builtin(__builtin_amdgcn_s_wait_asynccnt)
  __builtin_amdgcn_s_wait_asynccnt(0);
#else
  asm volatile("s_wait_asynccnt 0" ::: "memory");
#endif
#endif
}

// ---------------------------------------------------------------------------
// Kernel 1: fold (gamma+1)*D^-0.5 into a 2048x6 projection matrix.
// col 0..4 -> dynamic_alpha_fn, col 5 -> dynamic_beta_fn.
// ---------------------------------------------------------------------------
__global__ void build_bmod(const float* __restrict__ gamma,
                           const float* __restrict__ dyn_alpha,
                           const float* __restrict__ dyn_beta,
                           float* __restrict__ bmod) {
  int d = blockIdx.x * blockDim.x + threadIdx.x;
  if (d >= DDIM) return;
  float sc = (gamma[d] + 1.0f) * 0.02209708691207961f; // 2048^-0.5
  #pragma unroll
  for (int e = 0; e < 5; ++e) bmod[d * 6 + e] = sc * dyn_alpha[d * 5 + e];
  bmod[d * 6 + 5] = sc * dyn_beta[d];
}

// ---------------------------------------------------------------------------
// Per-group compute: WMMA projection + norms -> Sinkhorn -> out = W^T @ R
// ---------------------------------------------------------------------------
__device__ __forceinline__ void compute_group(
    int g, const float* Rbuf, const float* Bop,
    float* Cbuf, float* SSp, float* Csum, float* Inv, float* Wm,
    const float* __restrict__ static_alpha, const float* __restrict__ static_beta,
    const float* __restrict__ pre_scale, const float* __restrict__ res_scale,
    const float* __restrict__ beta_scale, float* __restrict__ out) {
  const int tid = threadIdx.x;
  const int lane = tid & 31;
  const int wv = tid >> 5;
  const int h = lane >> 4;                  // K half (0: K0/1, 1: K2/3)
  const int rn = lane & 15;                 // A row / B column
  const int bcol = (rn < 6) ? rn : 0;       // cols >=6 of C are never read

  // ---- WMMA: C(16x16) += A(16x4)*B(4x16); K split across 8 waves ----
  v8f acc = {};
  float ss = 0.0f;
  const int kbase = wv * KPW;
  for (int c = 0; c < CHUNKS; ++c) {
    int k0 = kbase + c * 4 + 2 * h;
    v2f a = *(const v2f*)&Rbuf[rn * DDIM + k0];   // ds_load_b64 (2addr-fused)
    v2f b;
    b.x = Bop[k0 * 6 + bcol];
    b.y = Bop[(k0 + 1) * 6 + bcol];
    ss += a.x * a.x + a.y * a.y;                  // L2 norm rides along
    acc = __builtin_amdgcn_wmma_f32_16x16x4_f32(
        false, a, false, b, (short)0, acc, false, false);
  }
  #pragma unroll
  for (int j = 0; j < 8; ++j) Cbuf[(wv * 8 + j) * 32 + lane] = acc[j];
  SSp[wv * 32 + lane] = ss;
  __syncthreads();

  // ---- Cross-wave reduction: 16x6 projections + 16 inverse norms ----
  if (tid < 96) {
    int m = tid / 6, n = tid % 6;
    float s = 0.0f;
    #pragma unroll
    for (int w = 0; w < NWAVES; ++w)
      s += Cbuf[(w * 8 + (m & 7)) * 32 + (n + 16 * (m >> 3))];
    Csum[m * 6 + n] = s;
  } else if (tid < 112) {
    int r = tid - 96;
    float s = 0.0f;
    #pragma unroll
    for (int w = 0; w < NWAVES; ++w)
      s += SSp[w * 32 + r] + SSp[w * 32 + 16 + r];
    Inv[r] = 1.0f / (sqrtf(s) + 1e-8f);
  }
  __syncthreads();

  // ---- Per-token: sigmoid / Sinkhorn(20) / fold into 4x4 mix W ----
  if (tid < TOK) {
    const int t = tid;
    const float psc = pre_scale[0], rsc = res_scale[0], bsc = beta_scale[0];
    float apre[4], bet[4], e[4][4];
    #pragma unroll
    for (int s = 0; s < 4; ++s) {
      int r = t * 4 + s;
      float inv = Inv[r];
      float a0 = Csum[r * 6 + 0] * inv * psc + static_alpha[s * 5 + 0];
      apre[s] = 1.0f / (1.0f + expf(-a0));
      #pragma unroll
      for (int j = 0; j < 4; ++j) {
        float aj = Csum[r * 6 + 1 + j] * inv * rsc + static_alpha[s * 5 + 1 + j];
        e[s][j] = expf(aj);
      }
      float dcv = Csum[r * 6 + 5] * inv * bsc + static_beta[s];
      bet[s] = 2.0f / (1.0f + expf(-dcv));
    }
    for (int it = 0; it < 20; ++it) {
      #pragma unroll
      for (int j = 0; j < 4; ++j) {   // axis=-2: column normalize
        float ic = 1.0f / (e[0][j] + e[1][j] + e[2][j] + e[3][j] + 1e-8f);
        #pragma unroll
        for (int s = 0; s < 4; ++s) e[s][j] *= ic;
      }
      #pragma unroll
      for (int s = 0; s < 4; ++s) {   // axis=-1: row normalize
        float ir = 1.0f / (e[s][0] + e[s][1] + e[s][2] + e[s][3] + 1e-8f);
        #pragma unroll
        for (int j = 0; j < 4; ++j) e[s][j] *= ir;
      }
    }
    #pragma unroll
    for (int s = 0; s < 4; ++s)
      #pragma unroll
      for (int j = 0; j < 4; ++j)
        Wm[t * 16 + s * 4 + j] = e[s][j] + apre[s] * bet[j];
  }
  __syncthreads();

  // ---- Output: out[j][d] = sum_s W[s][j]*R[s][d], streamed from LDS ----
  {
    const int t = tid >> 6;         // token (64 threads each)
    const int lp = tid & 63;
    float w[4][4];
    #pragma unroll
    for (int s = 0; s < 4; ++s)
      #pragma unroll
      for (int j = 0; j < 4; ++j) w[s][j] = Wm[t * 16 + s * 4 + j];
    float* dst = out + (size_t)g * RSIZE + (size_t)t * (SDIM * DDIM);
    const float* Rt = &Rbuf[t * SDIM * DDIM];
    #pragma unroll 2
    for (int i = 0; i < 8; ++i) {
      int d0 = lp * 4 + i * 256;
      f4 r0 = *(const f4*)&Rt[0 * DDIM + d0];
      f4 r1 = *(const f4*)&Rt[1 * DDIM + d0];
      f4 r2 = *(const f4*)&Rt[2 * DDIM + d0];
      f4 r3 = *(const f4*)&Rt[3 * DDIM + d0];
      #pragma unroll
      for (int j = 0; j < 4; ++j) {
        f4 o = r0 * w[0][j] + r1 * w[1][j] + r2 * w[2][j] + r3 * w[3][j];
        *(f4*)&dst[j * DDIM + d0] = o;   // global_store_b128
      }
    }
  }
}

// ---------------------------------------------------------------------------
// Kernel 2: persistent workgroup, double-buffered async DMA pipeline.
// ---------------------------------------------------------------------------
__global__ __launch_bounds__(NTHREADS) void mhc_main(
    const float* __restrict__ residuals,
    const float* __restrict__ static_alpha,
    const float* __restrict__ static_beta,
    const float* __restrict__ bmod,
    const float* __restrict__ pre_scale,
    const float* __restrict__ res_scale,
    const float* __restrict__ beta_scale,
    float* __restrict__ out) {
  __shared__ float R2[2][RSIZE];            // 2 x 128 KB double buffer
  __shared__ float Bop[DDIM * 6];           // 48 KB
  __shared__ float Cbuf[NWAVES * 8 * 32];   // 8 KB WMMA C partials
  __shared__ float SSp[NWAVES * 32];
  __shared__ float Csum[16 * 6];
  __shared__ float Inv[16];
  __shared__ float Wm[TOK * 16];

  const int tid = threadIdx.x;
  const int g0 = blockIdx.x * GPW;

  // ---- Prologue: stage Bmod + first R tile (async DMA) ----
  #pragma unroll 4
  for (int i = 0; i < RSIZE / (4 * NTHREADS); ++i) {   // 32 x b128 per thread
    int idx = (i * NTHREADS + tid) * 4;
    gload_lds128(residuals + (size_t)g0 * RSIZE + idx, &R2[0][idx]);
  }
  #pragma unroll 4
  for (int i = 0; i < (DDIM * 6) / (4 * NTHREADS); ++i) {  // 12 x b128
    int idx = (i * NTHREADS + tid) * 4;
    gload_lds128(bmod + idx, &Bop[idx]);
  }
  wait_async_lds();
  __syncthreads();

  // ---- Pipeline: prefetch group i+1 while computing group i ----
  for (int i = 0; i < GPW; ++i) {
    const int cur = i & 1;
    if (i + 1 < GPW) {
      const float* nsrc = residuals + (size_t)(g0 + i + 1) * RSIZE;
      #pragma unroll 4
      for (int k = 0; k < RSIZE / (4 * NTHREADS); ++k) {
        int idx = (k * NTHREADS + tid) * 4;
        gload_lds128(nsrc + idx, &R2[1 - cur][idx]);   // overlaps compute
      }
    }
    compute_group(g0 + i, R2[cur], Bop, Cbuf, SSp, Csum, Inv, Wm,
                  static_alpha, static_beta, pre_scale, res_scale, beta_scale,
                  out);
    wait_async_lds();
    __syncthreads();
  }
}

extern "C" void kernel_launch(void* const* d_in, const int* in_sizes, int n_in,
                              void* d_out, int out_size, void* d_ws, size_t ws_size,
                              hipStream_t stream) {
  const float* residuals    = (const float*)d_in[0];
  const float* gamma        = (const float*)d_in[1];
  const float* static_alpha = (const float*)d_in[2];
  const float* static_beta  = (const float*)d_in[3];
  const float* dyn_alpha    = (const float*)d_in[4];
  const float* dyn_beta     = (const float*)d_in[5];
  const float* pre_scale    = (const float*)d_in[6];
  const float* res_scale    = (const float*)d_in[7];
  const float* beta_scl     = (const float*)d_in[8];
  float* out  = (float*)d_out;
  float* bmod = (float*)d_ws;   // 2048*6 floats = 48KB scratch

  build_bmod<<<(DDIM + 255) / 256, 256, 0, stream>>>(gamma, dyn_alpha, dyn_beta, bmod);

  mhc_main<<<NWG, NTHREADS, 0, stream>>>(
      residuals, static_alpha, static_beta, bmod,
      pre_scale, res_scale, beta_scl, out);
}